// AttnLoss_26551487823965
// MI455X (gfx1250) — compile-verified
//
#include <hip/hip_runtime.h>
#include <hip/hip_bf16.h>

#define NCH 8
#define WW  64
#define HW  4096
#define SEQ 77

typedef __attribute__((ext_vector_type(2))) float v2f;
typedef __attribute__((ext_vector_type(8))) float v8f;

// ---- workspace layout (float element offsets) ----
#define WS_COLMAX 0                      // 8 floats : per-channel max of raw attn column
#define WS_MX     8                      // 8 floats : per-channel max of upsampled image
#define WS_PTOK   16                     // 8 floats : raw per-token sum of squared diffs
#define WS_LJ     64                     // 4096     : raw per-column sum of squared diffs
#define WS_GATES  (WS_LJ + HW)           // 8*4096   : gate (0/1) per channel per pixel
#define WS_PART   (WS_GATES + NCH * HW)  // 16*4096  : row-chunk partials for Lj

// Bilinear sample of the *scaled* channel image (255*a/colmax), 64->256 grid,
// align_corners=False semantics matching the reference exactly in structure.
__device__ __forceinline__ float bilerp_scaled(const float* __restrict__ attn,
                                               int col, float colmax, int r, int q) {
  float ys = fminf(fmaxf(((float)r + 0.5f) * 0.25f - 0.5f, 0.0f), 63.0f);
  float xs = fminf(fmaxf(((float)q + 0.5f) * 0.25f - 0.5f, 0.0f), 63.0f);
  int y0 = (int)floorf(ys), x0 = (int)floorf(xs);
  int y1 = min(y0 + 1, 63), x1 = min(x0 + 1, 63);
  float wy = ys - (float)y0, wx = xs - (float)x0;
  float s00 = (255.0f * attn[(y0 * WW + x0) * SEQ + col]) / colmax;
  float s01 = (255.0f * attn[(y0 * WW + x1) * SEQ + col]) / colmax;
  float s10 = (255.0f * attn[(y1 * WW + x0) * SEQ + col]) / colmax;
  float s11 = (255.0f * attn[(y1 * WW + x1) * SEQ + col]) / colmax;
  float top = s00 * (1.0f - wx) + s01 * wx;
  float bot = s10 * (1.0f - wx) + s11 * wx;
  return top * (1.0f - wy) + bot * wy;
}

// 1) per-channel max of raw attn column (tokens 1..8)
__global__ void k_colmax(const float* __restrict__ attn, float* __restrict__ ws) {
  int c = blockIdx.x, col = c + 1, t = threadIdx.x;
  float m = 0.0f;
  for (int i = t; i < HW; i += 256) m = fmaxf(m, attn[i * SEQ + col]);
  __shared__ float red[256];
  red[t] = m; __syncthreads();
  for (int s = 128; s > 0; s >>= 1) { if (t < s) red[t] = fmaxf(red[t], red[t + s]); __syncthreads(); }
  if (t == 0) ws[WS_COLMAX + c] = red[0];
}

// 2) per-token raw sum of squared diffs (cross-attn retain/erase)
__global__ void k_ptok(const float* __restrict__ attn, const float* __restrict__ attn_e,
                       float* __restrict__ ws) {
  int c = blockIdx.x, col = c + 1, t = threadIdx.x;
  float acc = 0.0f;
  for (int i = t; i < HW; i += 256) {
    float d = attn[i * SEQ + col] - attn_e[i * SEQ + col];
    acc += d * d;
  }
  __shared__ float red[256];
  red[t] = acc; __syncthreads();
  for (int s = 128; s > 0; s >>= 1) { if (t < s) red[t] += red[t + s]; __syncthreads(); }
  if (t == 0) ws[WS_PTOK + c] = red[0];
}

// 3) max over the full 256x256 upsampled image per channel
__global__ void k_mx(const float* __restrict__ attn, float* __restrict__ ws) {
  int c = blockIdx.x, col = c + 1, t = threadIdx.x;
  float colmax = ws[WS_COLMAX + c];
  float m = -3.0e38f;
  for (int p = t; p < 256 * 256; p += 256)
    m = fmaxf(m, bilerp_scaled(attn, col, colmax, p >> 8, p & 255));
  __shared__ float red[256];
  red[t] = m; __syncthreads();
  for (int s = 128; s > 0; s >>= 1) { if (t < s) red[t] = fmaxf(red[t], red[t + s]); __syncthreads(); }
  if (t == 0) ws[WS_MX + c] = red[0];
}

// 4) gates: mask>0  <=>  any of the 2x2 binary pixels {4o+1,4o+2}x{4p+1,4p+2} above mx*thr
__global__ void k_gates(const float* __restrict__ attn, float* __restrict__ ws) {
  int c = blockIdx.x;
  int j = blockIdx.y * 256 + threadIdx.x;   // 0..4095
  int o = j >> 6, p = j & 63;
  float colmax = ws[WS_COLMAX + c];
  float thr = (c == 1 || c == 4) ? 0.85f : 0.95f;   // token idx c+1 in {2,5} -> 0.85
  float lim = ws[WS_MX + c] * thr;
  int col = c + 1;
  bool g = false;
  #pragma unroll
  for (int dr = 1; dr <= 2; ++dr)
    #pragma unroll
    for (int dq = 1; dq <= 2; ++dq)
      g = g || (bilerp_scaled(attn, col, colmax, 4 * o + dr, 4 * p + dq) >= lim);
  ws[WS_GATES + c * HW + j] = g ? 1.0f : 0.0f;
}

// 5) Lj partials: each (bx,by) block sums 256 rows for 1024 columns (float4 streaming)
__global__ void k_lj_part(const float* __restrict__ sa, const float* __restrict__ se,
                          float* __restrict__ ws) {
  int j0 = blockIdx.x * 1024 + threadIdx.x * 4;
  int r0 = blockIdx.y * 256;
  float4 acc = make_float4(0.f, 0.f, 0.f, 0.f);
  for (int i = r0; i < r0 + 256; ++i) {
    const float4 a = *(const float4*)(sa + (size_t)i * HW + j0);
    const float4 b = *(const float4*)(se + (size_t)i * HW + j0);
    float dx = a.x - b.x, dy = a.y - b.y, dz = a.z - b.z, dw = a.w - b.w;
    acc.x += dx * dx; acc.y += dy * dy; acc.z += dz * dz; acc.w += dw * dw;
  }
  *(float4*)(ws + WS_PART + blockIdx.y * HW + j0) = acc;
}

// 6) deterministic fixed-order reduction of the 16 row-chunk partials
__global__ void k_lj_reduce(float* __restrict__ ws) {
  int j = blockIdx.x * 256 + threadIdx.x;
  float acc = 0.0f;
  #pragma unroll
  for (int r = 0; r < 16; ++r) acc += ws[WS_PART + r * HW + j];
  ws[WS_LJ + j] = acc;
}

// 7) final combine: per_i = Gates(8x4096) @ Lj via V_WMMA_F32_16X16X4_F32, one wave32.
// A 16x4 layout: VGPR0 = K{0|2}, VGPR1 = K{1|3} (split at lane 16), M = lane&15.
// B 4x16: Lj value broadcast across all N columns, same K split.
// D: lane 0 holds D[0..7][0] in acc[0..7] = per_i[0..7].
__global__ void k_final(const float* __restrict__ ws, float* __restrict__ out) {
  int lane = threadIdx.x;                 // 0..31
  int m = lane & 15;
  int hi = lane >> 4;                     // 0: K={0,1}, 1: K={2,3}
  int k0 = hi ? 2 : 0;
  int k1 = hi ? 3 : 1;
  float sel = (m < NCH) ? 1.0f : 0.0f;    // zero out rows 8..15 without divergence
  const float* grow = ws + WS_GATES + (size_t)((m < NCH) ? m : 0) * HW;
  const float* L = ws + WS_LJ;
  v8f acc = {0.f, 0.f, 0.f, 0.f, 0.f, 0.f, 0.f, 0.f};
  for (int j = 0; j < HW; j += 4) {
    v2f a, b;
    a[0] = sel * grow[j + k0];
    a[1] = sel * grow[j + k1];
    b[0] = L[j + k0];
    b[1] = L[j + k1];
    acc = __builtin_amdgcn_wmma_f32_16x16x4_f32(
        /*neg_a=*/false, a, /*neg_b=*/false, b,
        /*c_mod=*/(short)0, acc, /*reuse_a=*/false, /*reuse_b=*/false);
  }
  if (lane == 0) {
    const float sgn[NCH] = {1.f, -1.f, 1.f, 1.f, -1.f, 1.f, 1.f, 1.f};
    float tot = 0.0f;
    #pragma unroll
    for (int c = 0; c < NCH; ++c) tot += sgn[c] * (ws[WS_PTOK + c] + acc[c]);
    out[0] = tot * (1.0f / 4096.0f);      // exact power-of-two mean fold
  }
}

extern "C" void kernel_launch(void* const* d_in, const int* in_sizes, int n_in,
                              void* d_out, int out_size, void* d_ws, size_t ws_size,
                              hipStream_t stream) {
  (void)in_sizes; (void)n_in; (void)out_size; (void)ws_size;
  const float* attn   = (const float*)d_in[0];
  const float* attn_e = (const float*)d_in[1];
  const float* sa     = (const float*)d_in[2];
  const float* se     = (const float*)d_in[3];
  float* out = (float*)d_out;
  float* ws  = (float*)d_ws;

  k_colmax   <<<NCH, 256, 0, stream>>>(attn, ws);
  k_ptok     <<<NCH, 256, 0, stream>>>(attn, attn_e, ws);
  k_mx       <<<NCH, 256, 0, stream>>>(attn, ws);
  k_gates    <<<dim3(NCH, 16), 256, 0, stream>>>(attn, ws);
  k_lj_part  <<<dim3(4, 16), 256, 0, stream>>>(sa, se, ws);
  k_lj_reduce<<<16, 256, 0, stream>>>(ws);
  k_final    <<<1, 32, 0, stream>>>(ws, out);
}